// Tokenizer_81080392614191
// MI455X (gfx1250) — compile-verified
//
#include <hip/hip_runtime.h>

typedef __attribute__((ext_vector_type(2))) float v2f;
typedef __attribute__((ext_vector_type(8))) float v8f;

#define EPSV   1e-6f
#define C_DIM  256
#define T_DIM  2048
#define N_TOK  32768
#define V_CODES 1024
#define B_STRIDE 260   // 256 + 4 dword pad -> stride mod 64 == 4 (conflict-free b64 reads)

__device__ __forceinline__ float waveRedSum(float v) {
#pragma unroll
  for (int off = 16; off >= 1; off >>= 1)
    v += __shfl_xor(v, off, 32);
  return v;
}

// Async global->LDS copy, 16 bytes per lane (CDNA5 gfx1250).
__device__ __forceinline__ void async_load_b128(unsigned lds_off, const void* gptr) {
  asm volatile("global_load_async_to_lds_b128 %0, %1, off"
               :: "v"(lds_off), "v"(gptr)
               : "memory");
}
__device__ __forceinline__ void wait_async0() {
  asm volatile("s_wait_asynccnt 0x0" ::: "memory");
}

// ---------------------------------------------------------------------------
// Kernel 1: nearest-code search via fp32 WMMA, async-LDS-staged codebook tiles.
// Block = 8 waves = 128 tokens. All 8 waves share each 16-code B tile from LDS.
// ---------------------------------------------------------------------------
__global__ __launch_bounds__(256) void vq_argmax_kernel(
    const float* __restrict__ z,   // (N_TOK, 256)
    const float* __restrict__ cb,  // (1025, 256); codes are rows 1..1024
    int* __restrict__ best_idx)    // (N_TOK,)
{
  __shared__ float smem[2][16 * B_STRIDE];   // double-buffered B tile, row-padded

  const int lane = threadIdx.x & 31;
  const int wave = threadIdx.x >> 5;
  const int rowBase = (blockIdx.x * 8 + wave) * 16;
  const int half = lane >> 4;   // 0: lanes 0-15 (K=0,1), 1: lanes 16-31 (K=2,3)
  const int l16  = lane & 15;

  // A-matrix panel resident in VGPRs: a[k] holds K = 4k+2*half .. +1 of row M=l16.
  v2f a[64];
  const float* zrow = z + (size_t)(rowBase + l16) * C_DIM + half * 2;
#pragma unroll
  for (int k = 0; k < 64; ++k)
    a[k] = *(const v2f*)(zrow + k * 4);

  float bestVal[8];
  int   bestCode[8];
#pragma unroll
  for (int j = 0; j < 8; ++j) { bestVal[j] = -3.0e38f; bestCode[j] = 0; }

  const char* ebase = (const char*)(cb + C_DIM);  // skip codebook row 0

  // Wave `wave` stages rows 2*wave, 2*wave+1 of a tile: 4 x (32 lanes x 16B).
  auto issue_tile_loads = [&](int vt, int b) {
    const char* gsrc = ebase + (size_t)(vt * 16) * (C_DIM * 4);
#pragma unroll
    for (int i = 0; i < 4; ++i) {
      const int r = 2 * wave + (i >> 1);
      const int c = i & 1;
      unsigned loff = (unsigned)(size_t)&smem[b][r * B_STRIDE] + c * 512 + lane * 16;
      const char* g = gsrc + r * (C_DIM * 4) + c * 512 + lane * 16;
      async_load_b128(loff, g);
    }
  };

  issue_tile_loads(0, 0);
  wait_async0();
  __syncthreads();

  for (int vt = 0; vt < 64; ++vt) {
    const int b = vt & 1;
    if (vt < 63) issue_tile_loads(vt + 1, b ^ 1);   // prefetch next tile (uniform)

    // Consume tile b from LDS: lane reads code row l16, banks 4*l16 + 2*half.
    const float* bb = &smem[b][l16 * B_STRIDE + half * 2];
    v8f acc0 = {}, acc1 = {};
#pragma unroll
    for (int k = 0; k < 64; k += 2) {
      v2f b0 = *(const v2f*)(bb + k * 4);
      v2f b1 = *(const v2f*)(bb + (k + 1) * 4);
      acc0 = __builtin_amdgcn_wmma_f32_16x16x4_f32(false, a[k],     false, b0,
                                                   (short)0, acc0, false, false);
      acc1 = __builtin_amdgcn_wmma_f32_16x16x4_f32(false, a[k + 1], false, b1,
                                                   (short)0, acc1, false, false);
    }
    v8f accs = acc0 + acc1;

    // accs[j] = dot(row j + 8*half, code vt*16 + l16). argmax, low-index ties.
    const int code = vt * 16 + l16;
#pragma unroll
    for (int j = 0; j < 8; ++j) {
      float v = accs[j];
      bool take = (v > bestVal[j]) | ((v == bestVal[j]) & (code < bestCode[j]));
      bestVal[j]  = take ? v    : bestVal[j];
      bestCode[j] = take ? code : bestCode[j];
    }

    wait_async0();      // next tile fully in LDS (this wave's share)
    __syncthreads();    // all waves' shares visible before next iteration
  }

  // Reduce (value,index) across the 16 lanes of each half-wave.
#pragma unroll
  for (int off = 8; off >= 1; off >>= 1) {
#pragma unroll
    for (int j = 0; j < 8; ++j) {
      float ov = __shfl_xor(bestVal[j], off, 32);
      int   oc = __shfl_xor(bestCode[j], off, 32);
      bool take = (ov > bestVal[j]) | ((ov == bestVal[j]) & (oc < bestCode[j]));
      bestVal[j]  = take ? ov : bestVal[j];
      bestCode[j] = take ? oc : bestCode[j];
    }
  }
  if (l16 == 0) {
#pragma unroll
    for (int j = 0; j < 8; ++j)
      best_idx[rowBase + half * 8 + j] = bestCode[j];
  }
}

// ---------------------------------------------------------------------------
// Kernel 2: per-token epilogue (gather, rotation trick, losses). One wave/token.
// ---------------------------------------------------------------------------
__global__ __launch_bounds__(256) void vq_epilogue_kernel(
    const float* __restrict__ z, const float* __restrict__ cb,
    const float* __restrict__ mask, const int* __restrict__ best_idx,
    float* __restrict__ zq_out,     // d_out + 2, (N_TOK,256)
    float* __restrict__ partials)   // (gridDim.x, 2): commit, smooth
{
  __shared__ float s_commit[8];
  __shared__ float s_smooth[8];
  const int lane = threadIdx.x & 31;
  const int wave = threadIdx.x >> 5;
  const int n = blockIdx.x * 8 + wave;
  const int t = n & (T_DIM - 1);

  const float  m    = mask[n];
  const int    code = best_idx[n];
  const float* xrow = z + (size_t)n * C_DIM;
  const float* erow = cb + C_DIM + (size_t)code * C_DIM;

  float x[8], q[8];
  float xx = 0.f, commit = 0.f;
#pragma unroll
  for (int i = 0; i < 8; ++i) {
    const int d = i * 32 + lane;
    float xv = xrow[d];
    float ev = erow[d];          // z_q before ST/mask
    x[i] = xv;
    q[i] = ev * m;               // z_q after ST (==e[idx]) then * mask
    xx += xv * xv;
    float df = xv - ev;          // commitment uses unmasked z_q
    commit += df * df;
  }
  xx     = waveRedSum(xx);
  commit = waveRedSum(commit) * m;

  const float xn  = sqrtf(xx);
  const float inv = 1.0f / (xn + EPSV);

  float s[8];
  float ss = 0.f;
#pragma unroll
  for (int i = 0; i < 8; ++i) {
    float u = x[i] * inv;
    s[i] = u + q[i];
    ss += s[i] * s[i];
  }
  ss = waveRedSum(ss);
  const float winv = 1.0f / sqrtf(ss);   // reference: no eps in ||u+q||

  float dot_w = 0.f, dot_u = 0.f;
#pragma unroll
  for (int i = 0; i < 8; ++i) {
    float w = s[i] * winv;
    s[i] = w;
    dot_w += x[i] * w;
    dot_u += x[i] * (x[i] * inv);
  }
  dot_w = waveRedSum(dot_w);
  dot_u = waveRedSum(dot_u);

  float* orow = zq_out + (size_t)n * C_DIM;
#pragma unroll
  for (int i = 0; i < 8; ++i) {
    const int d = i * 32 + lane;
    float pre = x[i] - 2.f * dot_w * s[i] + 2.f * dot_u * q[i];
    orow[d] = pre * (q[i] * inv) * m;    // elementwise scale sg(zq/(|x|+eps)), then *mask
  }

  // Smoothness: pair (t, t+1) weighted by mask[t+1]; branch is wave-uniform.
  float smooth = 0.f;
  if (t < T_DIM - 1) {
    const float* xnrow = z + (size_t)(n + 1) * C_DIM;
    const float mn = mask[n + 1];
    float acc = 0.f;
#pragma unroll
    for (int i = 0; i < 8; ++i) {
      const int d = i * 32 + lane;
      float df = x[i] - xnrow[d];
      acc += df * df;
    }
    smooth = waveRedSum(acc) * mn;
  }

  if (lane == 0) { s_commit[wave] = commit; s_smooth[wave] = smooth; }
  __syncthreads();
  if (threadIdx.x == 0) {
    float c = 0.f, sm = 0.f;
#pragma unroll
    for (int wv = 0; wv < 8; ++wv) { c += s_commit[wv]; sm += s_smooth[wv]; }
    partials[blockIdx.x * 2 + 0] = c;
    partials[blockIdx.x * 2 + 1] = sm;
  }
}

// ---------------------------------------------------------------------------
// Kernel 3: deterministic final reduction of scalars.
// ---------------------------------------------------------------------------
__global__ __launch_bounds__(256) void vq_reduce_kernel(
    const float* __restrict__ partials, int nPart,
    const float* __restrict__ mask, float* __restrict__ out01)
{
  __shared__ float sc[256], ssm[256], sv[256];
  const int t = threadIdx.x;
  float c = 0.f, sm = 0.f, v = 0.f;
  for (int i = t; i < nPart; i += 256) {
    c  += partials[i * 2 + 0];
    sm += partials[i * 2 + 1];
  }
  for (int i = t; i < N_TOK; i += 256) v += mask[i];
  sc[t] = c; ssm[t] = sm; sv[t] = v;
  __syncthreads();
  for (int off = 128; off >= 1; off >>= 1) {
    if (t < off) { sc[t] += sc[t + off]; ssm[t] += ssm[t + off]; sv[t] += sv[t + off]; }
    __syncthreads();
  }
  if (t == 0) {
    const float valid = sv[0];
    out01[0] = ssm[0] / valid;  // smoothness_loss
    out01[1] = sc[0]  / valid;  // commitment_loss
  }
}

extern "C" void kernel_launch(void* const* d_in, const int* in_sizes, int n_in,
                              void* d_out, int out_size, void* d_ws, size_t ws_size,
                              hipStream_t stream) {
  const float* z    = (const float*)d_in[0];
  const float* cb   = (const float*)d_in[1];
  const float* mask = (const float*)d_in[2];
  float* out = (float*)d_out;

  int*   best_idx = (int*)d_ws;                                  // 128 KB
  float* partials = (float*)((char*)d_ws + N_TOK * sizeof(int)); // 32 KB

  const int nBlocksEpi = N_TOK / 8;  // 4096 blocks, 1 wave per token
  vq_argmax_kernel  <<<N_TOK / 128, 256, 0, stream>>>(z, cb, best_idx);
  vq_epilogue_kernel<<<nBlocksEpi, 256, 0, stream>>>(z, cb, mask, best_idx,
                                                     out + 2, partials);
  vq_reduce_kernel  <<<1, 256, 0, stream>>>(partials, nBlocksEpi, mask, out);
}